// ShortDistanceAttention_34969623724581
// MI455X (gfx1250) — compile-verified
//
#include <hip/hip_runtime.h>
#include <math.h>

// ---------------------------------------------------------------------------
// ShortDistanceAttention on MI455X (gfx1250, wave32)
//   out = gelu( att_k @ Wh ),  att_k = attention * (A^2 != 0)
// Strategy:
//   1) bit-pack A                       (one 268 MB streamed read, ~12us @23.3TB/s)
//   2) s_src/s_dst + rowsum + A^2 mask  (bitset union over ~33 nbrs/row; replaces
//                                        the 1.1 TFLOP dense A@A with ~270 MB of OR)
//   3) fused attention-gen + fp32 WMMA GEMM + GELU, double-buffered via
//      GLOBAL_LOAD_ASYNC_TO_LDS_B128 (ASYNCcnt) when the toolchain exposes it.
// Matrix ops use V_WMMA_F32_16X16X4_F32 (fp32, no precision loss).
// ---------------------------------------------------------------------------

#define N_NODES 8192
#define IN_F    256
#define OUT_F   128
#define ALPHA   0.2f
#define NWORDS  (N_NODES / 32)     // 256 u32 words per bitmask row
#define KC      64                 // K-chunk for fused GEMM
#define KCP     (KC + 1)           // padded LDS stride (bank-conflict free)
#define MAXNBR  2048

typedef float v2f __attribute__((ext_vector_type(2)));
typedef float v8f __attribute__((ext_vector_type(8)));
typedef int   v4i __attribute__((ext_vector_type(4)));
typedef __attribute__((address_space(1))) v4i g_v4i;   // global-AS int4 pointee
typedef __attribute__((address_space(3))) v4i l_v4i;   // LDS-AS int4 pointee

#if defined(__has_builtin)
#  if __has_builtin(__builtin_amdgcn_global_load_async_to_lds_b128)
#    define HAS_ASYNC_LDS 1
#  endif
#  if __has_builtin(__builtin_amdgcn_s_wait_asynccnt)
#    define HAS_WAIT_ASYNC_BUILTIN 1
#  endif
#endif

__device__ __forceinline__ void async_wait0() {
#if defined(HAS_ASYNC_LDS)
#  if defined(HAS_WAIT_ASYNC_BUILTIN)
    __builtin_amdgcn_s_wait_asynccnt(0);
#  else
    asm volatile("s_wait_asynccnt 0x0" ::: "memory");
#  endif
#endif
}

// Stage KC*OUT_F floats (32 KB) of Wh into an LDS buffer.
// Async path: per-lane GLOBAL_LOAD_ASYNC_TO_LDS_B128, no VGPR staging.
__device__ __forceinline__ void stage_wh(const float4* __restrict__ gsrc,
                                         float4* ldst, int tid) {
#if defined(HAS_ASYNC_LDS)
    #pragma unroll
    for (int k = 0; k < (KC * OUT_F / 4) / 256; ++k) {
        const int idx = tid + k * 256;
        __builtin_amdgcn_global_load_async_to_lds_b128(
            (g_v4i*)(gsrc + idx),
            (l_v4i*)(ldst + idx),
            0, 0);
    }
#else
    #pragma unroll
    for (int k = 0; k < (KC * OUT_F / 4) / 256; ++k) {
        const int idx = tid + k * 256;
        ldst[idx] = gsrc[idx];
    }
#endif
}

// Generate one 16 x KC attention tile into LDS (this thread's 4 entries).
__device__ __forceinline__ void gen_att_tile(float* attbuf, int j0,
                                             int m_att, int jj0, int row_att,
                                             float si, float ri,
                                             const unsigned* __restrict__ Abits,
                                             const unsigned* __restrict__ Akbits,
                                             const float* __restrict__ sdst) {
    const int j = j0 + jj0;                 // jj0 multiple of 4 -> same u32 word
    const unsigned akw = Akbits[(size_t)row_att * NWORDS + (j >> 5)];
    const unsigned abw = Abits [(size_t)row_att * NWORDS + (j >> 5)];
    #pragma unroll
    for (int q = 0; q < 4; ++q) {
        const int jq = j + q;
        float val = 0.f;
        if ((akw >> (jq & 31)) & 1u) {
            if ((abw >> (jq & 31)) & 1u) {
                float e = si + sdst[jq];
                e = (e > 0.f) ? e : (ALPHA * e);
                val = expf(e) * ri;
            } else {
                val = ri;                   // non-edge 2-hop entry: exp-mask = 1
            }
        }
        attbuf[m_att * KCP + jj0 + q] = val;
    }
}

// ---------------------------------------------------------------------------
// K1: pack adjacency row bits: Abits[row][w] bit b = (A[row][w*32+b] != 0)
// ---------------------------------------------------------------------------
__global__ __launch_bounds__(256) void sda_pack_bits(const float* __restrict__ A,
                                                     unsigned* __restrict__ Abits) {
    int t   = blockIdx.x * blockDim.x + threadIdx.x;   // N_NODES*NWORDS threads
    int row = t >> 8;
    int w   = t & (NWORDS - 1);
    const float4* p = (const float4*)(A + (size_t)row * N_NODES + (size_t)w * 32);
    unsigned word = 0u;
    #pragma unroll
    for (int q = 0; q < 8; ++q) {
        float4 v = p[q];
        word |= (v.x != 0.0f) ? (1u << (q * 4 + 0)) : 0u;
        word |= (v.y != 0.0f) ? (1u << (q * 4 + 1)) : 0u;
        word |= (v.z != 0.0f) ? (1u << (q * 4 + 2)) : 0u;
        word |= (v.w != 0.0f) ? (1u << (q * 4 + 3)) : 0u;
    }
    Abits[(size_t)row * NWORDS + w] = word;
}

// ---------------------------------------------------------------------------
// K2: Wh = X @ W via V_WMMA_F32_16X16X4_F32.
// Block = 8 waves; block owns 16 rows, wave w owns cols [16w,16w+16).
// fp32 fragment layout (ISA 7.12.2): A: lane l -> M=l&15, VGPR v -> K=v+2*(l>>4)
// (VGPR0={K0|K2}, VGPR1={K1|K3}); B symmetric; C: VGPR v -> M=v+8*(l>>4), N=l&15.
// ---------------------------------------------------------------------------
__global__ __launch_bounds__(256) void sda_wh_gemm(const float* __restrict__ X,
                                                   const float* __restrict__ W,
                                                   float* __restrict__ Wh) {
    const int rb   = blockIdx.x * 16;
    const int wave = threadIdx.x >> 5;
    const int lane = threadIdx.x & 31;
    const int half = lane >> 4;
    const int mn   = lane & 15;
    const int cb   = wave * 16;

    v8f c = {};
    for (int k0 = 0; k0 < IN_F; k0 += 4) {
        const int ka = k0 + 2 * half;
        v2f a, b;
        a.x = X[(size_t)(rb + mn) * IN_F + ka];
        a.y = X[(size_t)(rb + mn) * IN_F + ka + 1];
        b.x = W[(size_t)(ka)     * OUT_F + cb + mn];
        b.y = W[(size_t)(ka + 1) * OUT_F + cb + mn];
        c = __builtin_amdgcn_wmma_f32_16x16x4_f32(false, a, false, b,
                                                  (short)0, c, false, false);
    }
    #pragma unroll
    for (int v = 0; v < 8; ++v)
        Wh[(size_t)(rb + v + 8 * half) * OUT_F + cb + mn] = c[v];
}

// ---------------------------------------------------------------------------
// K3: per-node scores s_src = Wh @ r[:128], s_dst = Wh @ r[128:]
// ---------------------------------------------------------------------------
__global__ __launch_bounds__(256) void sda_scores(const float* __restrict__ Wh,
                                                  const float* __restrict__ r,
                                                  float* __restrict__ ssrc,
                                                  float* __restrict__ sdst) {
    int i = blockIdx.x * blockDim.x + threadIdx.x;
    if (i >= N_NODES) return;
    float a = 0.f, b = 0.f;
    #pragma unroll 8
    for (int d = 0; d < OUT_F; ++d) {
        float w = Wh[(size_t)i * OUT_F + d];
        a += w * r[d];
        b += w * r[OUT_F + d];
    }
    ssrc[i] = a;
    sdst[i] = b;
}

// ---------------------------------------------------------------------------
// K4: per row i (one 256-thread block):
//   - neighbor scan of Abits[i] -> rowinv[i] = 1 / sum_{edges} exp(lrelu(...))
//   - Akbits[i] = OR over neighbors j of Abits[j]   (A has self loops, so
//     i is in its own neighbor list => edges subseteq 2-hop mask)
// Thread t owns bitmask word t (coalesced row reads in the union loop).
// ---------------------------------------------------------------------------
__global__ __launch_bounds__(256) void sda_hop_rowsum(const unsigned* __restrict__ Abits,
                                                      const float* __restrict__ ssrc,
                                                      const float* __restrict__ sdst,
                                                      unsigned* __restrict__ Akbits,
                                                      float* __restrict__ rowinv) {
    __shared__ int   s_nbr[MAXNBR];
    __shared__ int   s_cnt;
    __shared__ float s_red[8];

    const int row = blockIdx.x;
    const int t   = threadIdx.x;
    if (t == 0) s_cnt = 0;
    __syncthreads();

    const unsigned aw = Abits[(size_t)row * NWORDS + t];
    const float    si = ssrc[row];

    float local = 0.f;
    unsigned w = aw;
    while (w) {
        int b = __ffs(w) - 1;
        w &= (w - 1);
        int j = t * 32 + b;
        int p = atomicAdd(&s_cnt, 1);
        if (p < MAXNBR) s_nbr[p] = j;
        float e = si + sdst[j];
        e = (e > 0.f) ? e : (ALPHA * e);
        local += expf(e);
    }
    // wave32 reduction, then cross-wave via LDS
    #pragma unroll
    for (int o = 16; o > 0; o >>= 1) local += __shfl_down(local, o, 32);
    if ((t & 31) == 0) s_red[t >> 5] = local;
    __syncthreads();                         // finalizes s_cnt, s_nbr, s_red
    if (t == 0) {
        float s = 0.f;
        #pragma unroll
        for (int k = 0; k < 8; ++k) s += s_red[k];
        rowinv[row] = 1.0f / s;
    }
    const int C = (s_cnt < MAXNBR) ? s_cnt : MAXNBR;
    unsigned acc = 0u;
    for (int n = 0; n < C; ++n) {
        int j = s_nbr[n];
        acc |= Abits[(size_t)j * NWORDS + t];
    }
    Akbits[(size_t)row * NWORDS + t] = acc;
}

// ---------------------------------------------------------------------------
// K5: out = gelu( att_k @ Wh ), fused, double-buffered.
// Block owns 16 output rows x all 128 cols (8 waves x one 16x16 tile each).
// Pipeline per K-chunk of 64:
//   issue async Wh loads for chunk c+1 -> LDS buf^1 (ASYNCcnt path),
//   generate att tile c+1 -> LDS buf^1 (exp only on true edges),
//   16 x V_WMMA_F32_16X16X4_F32 on chunk c from LDS buf,
//   s_wait_asynccnt 0 + barrier, swap.
// ---------------------------------------------------------------------------
__global__ __launch_bounds__(256) void sda_attn_gemm(const float* __restrict__ Wh,
                                                     const unsigned* __restrict__ Abits,
                                                     const unsigned* __restrict__ Akbits,
                                                     const float* __restrict__ ssrc,
                                                     const float* __restrict__ sdst,
                                                     const float* __restrict__ rowinv,
                                                     float* __restrict__ out) {
    __shared__ float sWh [2][KC * OUT_F];   // 2 x 32 KB
    __shared__ float sAtt[2][16 * KCP];     // 2 x 4.2 KB, padded stride

    const int rb   = blockIdx.x * 16;
    const int tid  = threadIdx.x;
    const int wave = tid >> 5;
    const int lane = tid & 31;
    const int half = lane >> 4;
    const int mn   = lane & 15;
    const int cb   = wave * 16;

    // att-gen assignment: thread covers 4 consecutive j's of one row
    const int m_att   = (tid * 4) >> 6;     // 0..15
    const int jj0     = (tid * 4) & 63;     // multiple of 4 (same u32 word)
    const int row_att = rb + m_att;
    const float si = ssrc[row_att];
    const float ri = rowinv[row_att];

    const int NCH = N_NODES / KC;           // 128 chunks

    // prologue: fill buffer 0
    stage_wh((const float4*)Wh, (float4*)sWh[0], tid);
    gen_att_tile(sAtt[0], 0, m_att, jj0, row_att, si, ri, Abits, Akbits, sdst);
    async_wait0();
    __syncthreads();

    v8f c = {};
    for (int ch = 0; ch < NCH; ++ch) {
        const int cur = ch & 1;
        if (ch + 1 < NCH) {
            stage_wh((const float4*)(Wh + (size_t)(ch + 1) * KC * OUT_F),
                     (float4*)sWh[cur ^ 1], tid);
            gen_att_tile(sAtt[cur ^ 1], (ch + 1) * KC, m_att, jj0, row_att,
                         si, ri, Abits, Akbits, sdst);
        }
        const float* att = sAtt[cur];
        const float* whb = sWh[cur];
        #pragma unroll
        for (int s = 0; s < KC; s += 4) {
            const int ka = s + 2 * half;
            v2f a, b;
            a.x = att[mn * KCP + ka];
            a.y = att[mn * KCP + ka + 1];
            b.x = whb[(ka)     * OUT_F + cb + mn];
            b.y = whb[(ka + 1) * OUT_F + cb + mn];
            c = __builtin_amdgcn_wmma_f32_16x16x4_f32(false, a, false, b,
                                                      (short)0, c, false, false);
        }
        async_wait0();
        __syncthreads();
    }

    // exact GELU + store
    #pragma unroll
    for (int v = 0; v < 8; ++v) {
        float x = c[v];
        float g = 0.5f * x * (1.0f + erff(x * 0.70710678118654752f));
        out[(size_t)(rb + v + 8 * half) * OUT_F + cb + mn] = g;
    }
}

// ---------------------------------------------------------------------------
// launch
// ---------------------------------------------------------------------------
extern "C" void kernel_launch(void* const* d_in, const int* in_sizes, int n_in,
                              void* d_out, int out_size, void* d_ws, size_t ws_size,
                              hipStream_t stream) {
    const float* X = (const float*)d_in[0];   // (8192, 256)
    const float* A = (const float*)d_in[1];   // (8192, 8192)
    const float* W = (const float*)d_in[2];   // (256, 128)
    const float* r = (const float*)d_in[3];   // (256, 1)
    float* out = (float*)d_out;               // (8192, 128)

    // workspace carve-up (~20.1 MB)
    char* ws = (char*)d_ws;
    float* Wh     = (float*)ws;  ws += (size_t)N_NODES * OUT_F * sizeof(float);
    float* ssrc   = (float*)ws;  ws += (size_t)N_NODES * sizeof(float);
    float* sdst   = (float*)ws;  ws += (size_t)N_NODES * sizeof(float);
    float* rowinv = (float*)ws;  ws += (size_t)N_NODES * sizeof(float);
    unsigned* Abits  = (unsigned*)ws;  ws += (size_t)N_NODES * NWORDS * sizeof(unsigned);
    unsigned* Akbits = (unsigned*)ws;

    sda_pack_bits <<<(N_NODES * NWORDS) / 256, 256, 0, stream>>>(A, Abits);
    sda_wh_gemm   <<<N_NODES / 16,            256, 0, stream>>>(X, W, Wh);
    sda_scores    <<<N_NODES / 256,           256, 0, stream>>>(Wh, r, ssrc, sdst);
    sda_hop_rowsum<<<N_NODES,                 256, 0, stream>>>(Abits, ssrc, sdst,
                                                               Akbits, rowinv);
    sda_attn_gemm <<<N_NODES / 16,            256, 0, stream>>>(Wh, Abits, Akbits,
                                                               ssrc, sdst, rowinv, out);
}